// TensorNeRF_50749333570317
// MI455X (gfx1250) — compile-verified
//
#include <hip/hip_runtime.h>
#include <stdint.h>

// ---------------------------------------------------------------------------
// NeRF volume-rendering weights: alpha / cumprod-transmittance / bg weight.
// Memory-bound (512 MB @ 23.3 TB/s => ~22 us floor). One wave32 per ray,
// 4 samples per lane per chunk, multiplicative Hillis-Steele wave scan,
// double-buffered GLOBAL_LOAD_ASYNC_TO_LDS_B128 pipeline (ASYNCcnt),
// nontemporal B128 stores.
// ---------------------------------------------------------------------------

#define NS              512            // samples per ray (reference-fixed)
#define CHUNK           128            // samples per wave iteration (4/lane)
#define NCHUNK          (NS / CHUNK)   // 4
#define WAVES_PER_BLOCK 8
#define BLOCK           (WAVES_PER_BLOCK * 32)

typedef float f4 __attribute__((ext_vector_type(4)));

// Async DMA of 16 bytes/lane from global memory into LDS.
// VDST = per-lane LDS byte offset (workgroup-relative), VADDR = 64-bit global addr.
__device__ __forceinline__ void async_b128(uint32_t lds_off, const float* g) {
    asm volatile("global_load_async_to_lds_b128 %0, %1, off"
                 :
                 : "v"(lds_off), "v"(g)
                 : "memory");
}

__global__ __launch_bounds__(BLOCK) void nerf_vr_scan(
    const float* __restrict__ sigma,
    const float* __restrict__ dist,
    float* __restrict__ alpha_o,
    float* __restrict__ w_o,
    float* __restrict__ bg_o,
    int nrays)
{
    // Per wave: 2 buffers x (128 sigma + 128 dist) floats = 2 KB.
    __shared__ __align__(16) float lds[WAVES_PER_BLOCK * 2 * 2 * CHUNK];

    const int lane = threadIdx.x & 31;
    const int wid  = threadIdx.x >> 5;
    const int ray  = (int)((blockIdx.x * BLOCK + threadIdx.x) >> 5);
    if (ray >= nrays) return;   // wave-uniform

    float* wls = &lds[wid * (2 * 2 * CHUNK)];
    const float* srow = sigma   + (size_t)ray * NS;
    const float* drow = dist    + (size_t)ray * NS;
    float*       arow = alpha_o + (size_t)ray * NS;
    float*       wrow = w_o     + (size_t)ray * NS;

    auto issue = [&](int c) {
        const int b = c & 1;
        float* sb = wls + b * (2 * CHUNK);
        const int col = c * CHUNK + lane * 4;
        async_b128((uint32_t)(uintptr_t)(sb + lane * 4),         srow + col);
        async_b128((uint32_t)(uintptr_t)(sb + CHUNK + lane * 4), drow + col);
    };

    issue(0);                       // prologue: chunk 0 in flight (cnt=2)

    float carry = 1.0f;             // running transmittance T across chunks
#pragma unroll
    for (int c = 0; c < NCHUNK; ++c) {
        const int b = c & 1;
        if (c + 1 < NCHUNK) {
            issue(c + 1);                               // cnt -> 4
            __builtin_amdgcn_s_wait_asynccnt(2);        // chunk c landed
        } else {
            __builtin_amdgcn_s_wait_asynccnt(0);        // last chunk landed
        }
        asm volatile("" ::: "memory");                  // no hoisting of LDS reads

        const float* sb = wls + b * (2 * CHUNK);
        f4 s4 = *(const f4*)(sb + lane * 4);
        f4 d4 = *(const f4*)(sb + CHUNK + lane * 4);

        float a[4], t[4];
        float p = 1.0f;                                 // lane-local 4-product
#pragma unroll
        for (int i = 0; i < 4; ++i) {
            a[i] = 1.0f - __expf(-s4[i] * d4[i]);       // alpha
            t[i] = 1.0f - a[i] + 1e-10f;                // survival factor
            p *= t[i];
        }

        // Inclusive multiplicative scan of p across the wave (5 shuffle steps).
        float P = p;
#pragma unroll
        for (unsigned off = 1; off < 32; off <<= 1) {
            float n = __shfl_up(P, off, 32);
            if (lane >= (int)off) P *= n;
        }
        // Exclusive prefix for this lane, threaded through the carry.
        float ex  = __shfl_up(P, 1, 32);
        float run = carry * ((lane == 0) ? 1.0f : ex);  // T at first sample of lane

        float w[4];
#pragma unroll
        for (int i = 0; i < 4; ++i) {
            w[i] = a[i] * run;                          // weight = alpha * T_prefix
            run *= t[i];
        }
        carry *= __shfl(P, 31, 32);                     // fold whole chunk into carry

        const int col = c * CHUNK + lane * 4;
        f4 av = {a[0], a[1], a[2], a[3]};
        f4 wv = {w[0], w[1], w[2], w[3]};
        __builtin_nontemporal_store(av, (f4*)(arow + col));
        __builtin_nontemporal_store(wv, (f4*)(wrow + col));
    }

    if (lane == 0) bg_o[ray] = carry;                   // T[:, -1]
}

extern "C" void kernel_launch(void* const* d_in, const int* in_sizes, int n_in,
                              void* d_out, int out_size, void* d_ws, size_t ws_size,
                              hipStream_t stream) {
    const float* sigma = (const float*)d_in[0];
    const float* dist  = (const float*)d_in[1];
    const int nrays = in_sizes[0] / NS;

    float* out   = (float*)d_out;
    const size_t n = (size_t)nrays * NS;
    float* alpha_o = out;            // [nrays*NS]
    float* w_o     = out + n;        // [nrays*NS]
    float* bg_o    = out + 2 * n;    // [nrays]

    const int blocks = (nrays + WAVES_PER_BLOCK - 1) / WAVES_PER_BLOCK;
    nerf_vr_scan<<<blocks, BLOCK, 0, stream>>>(sigma, dist, alpha_o, w_o, bg_o, nrays);
}